// PointChainPendulum2_8710193677068
// MI455X (gfx1250) — compile-verified
//
#include <hip/hip_runtime.h>

typedef __attribute__((ext_vector_type(2))) float v2f;
typedef __attribute__((ext_vector_type(8))) float v8f;

#define NPEND   128
#define NB      1024
#define LATENT  256
#define KT_STRIDE 258   // 256 + 2 pad (keeps 8B alignment, staggers banks)

// One chain (128 points) per workgroup; 256 threads = 8 wave32.
__global__ __launch_bounds__(256) void pend_lam_kernel(
    const float* __restrict__ y,    // [NB*NPEND, LATENT]
    const float* __restrict__ Kg,   // [LATENT, 6] row-major
    float* __restrict__ lam_y)      // [NB*NPEND, LATENT]
{
    __shared__ __align__(16) float KT[16][KT_STRIDE]; // KT[n][k] = K[k][n], n<3 else 0
    __shared__ __align__(16) float Kc[3][LATENT];     // Kc[d][l]  = K[l][d]
    __shared__ float4 r4[NPEND];                      // positions r (xyz, w unused)
    __shared__ float4 t4[NPEND];                      // t = c * dr / |dr|
    __shared__ float4 o4[NPEND];                      // out = t_i - t_{i+1}

    const int t     = threadIdx.x;
    const int lane  = t & 31;
    const int wave  = t >> 5;
    const int chain = blockIdx.x;

    // ---------------- Phase 0: stage K into LDS ----------------
    for (int idx = t; idx < 16 * LATENT; idx += 256) {
        int n = idx >> 8;          // padded x-column 0..15
        int k = idx & 255;         // latent index
        KT[n][k] = (n < 3) ? Kg[k * 6 + n] : 0.0f;
    }
    for (int idx = t; idx < 3 * LATENT; idx += 256) {
        int d = idx >> 8;
        int c = idx & 255;
        Kc[d][c] = Kg[c * 6 + d];
    }
    __syncthreads();

    // ---------------- Phase A: r = y_chain @ K[:, :3]  (WMMA f32 16x16x4) ----
    {
        const int m0   = wave * 16;            // 8 waves x 16 rows = 128 points
        const int mrow = m0 + (lane & 15);     // A: M = lane&15
        const int koff = (lane >> 4) * 2;      // A: upper half lanes hold K+2,K+3
        const float* yrow = y + (size_t)(chain * NPEND + mrow) * LATENT;

        v8f acc = {};
        for (int k = 0; k < LATENT; k += 4) {
            v2f a = *(const v2f*)(yrow + k + koff);                 // global_load_b64
            v2f b = *(const v2f*)(&KT[lane & 15][k + koff]);        // ds_load_b64
            acc = __builtin_amdgcn_wmma_f32_16x16x4_f32(
                      /*neg_a=*/false, a, /*neg_b=*/false, b,
                      /*c_mod=*/(short)0, acc,
                      /*reuse_a=*/false, /*reuse_b=*/false);
        }
        // D layout: vgpr v, lanes 0-15 -> M=m0+v, N=lane; lanes 16-31 -> M=m0+8+v
        int n = lane & 15;
        if (n < 3) {
            int   mbase = m0 + (lane >> 4) * 8;
            float* dst  = (float*)&r4[0];
            #pragma unroll
            for (int v = 0; v < 8; ++v)
                dst[(mbase + v) * 4 + n] = acc[v];
        }
    }
    __syncthreads();

    // ---------------- Phase B: chain constraint math ----------------
    if (t < NPEND) {
        float4 rp = r4[t];
        float dx, dy, dz;
        if (t == 0) { dx = rp.x; dy = rp.y; dz = rp.z; }
        else {
            float4 rm = r4[t - 1];
            dx = rp.x - rm.x; dy = rp.y - rm.y; dz = rp.z - rm.z;
        }
        float norm = sqrtf(dx * dx + dy * dy + dz * dz);
        float s    = (norm - 1.0f) / norm;      // c / |dr|,  D0 == 1.0
        t4[t] = make_float4(s * dx, s * dy, s * dz, 0.0f);
    }
    __syncthreads();
    if (t < NPEND) {
        float4 tp = t4[t];
        float4 tn = (t < NPEND - 1) ? t4[t + 1] : make_float4(0.f, 0.f, 0.f, 0.f);
        o4[t] = make_float4(tp.x - tn.x, tp.y - tn.y, tp.z - tn.z, 0.0f);
    }
    __syncthreads();

    // ---------------- Phase C: lam_y = out @ K[:, :3]^T  (rank-3, float4) ----
    {
        float* outg = lam_y + (size_t)chain * NPEND * LATENT;
        #pragma unroll 4
        for (int i = 0; i < 32; ++i) {
            int idx = i * 256 + t;        // 8192 float4s of the chain's output
            int p   = idx >> 6;           // point 0..127 (uniform per wave)
            int l   = (idx & 63) * 4;     // latent column (consec. across lanes)
            float4 o  = o4[p];                               // LDS broadcast
            float4 ka = *(const float4*)&Kc[0][l];           // ds_load_b128
            float4 kb = *(const float4*)&Kc[1][l];
            float4 kc = *(const float4*)&Kc[2][l];
            float4 v;
            v.x = o.x * ka.x + o.y * kb.x + o.z * kc.x;
            v.y = o.x * ka.y + o.y * kb.y + o.z * kc.y;
            v.z = o.x * ka.z + o.y * kb.z + o.z * kc.z;
            v.w = o.x * ka.w + o.y * kb.w + o.z * kc.w;
            *(float4*)(outg + p * LATENT + l) = v;           // global_store_b128
        }
    }
}

extern "C" void kernel_launch(void* const* d_in, const int* in_sizes, int n_in,
                              void* d_out, int out_size, void* d_ws, size_t ws_size,
                              hipStream_t stream) {
    const float* y  = (const float*)d_in[0];   // [N, 256]
    // d_in[1] = z   : unused by the reference math
    const float* Kg = (const float*)d_in[2];   // [256, 6]
    // d_in[3] = batch : deterministic (i // NPEND), structure baked in
    float* lam = (float*)d_out;                // [N, 256]
    pend_lam_kernel<<<NB, 256, 0, stream>>>(y, Kg, lam);
}